// GResBlock_19799799234725
// MI455X (gfx1250) — compile-verified
//
#include <hip/hip_runtime.h>

typedef float v2f __attribute__((ext_vector_type(2)));
typedef float v8f __attribute__((ext_vector_type(8)));

#define NNODES 20000
#define NEDGES 120000
#define CIN    963
#define CHID   256
#define COUT   3
#define NBATCH 4
#define NBLOCK 6

// ---------------------------------------------------------------------------
// fp32 WMMA GEMM: Out[M x CC] = A[M x K] (optionally row-gathered) * Bw[K x CC]
// One wave (32 threads) per (16*RT)-row x (16*CT)-col tile, using
// V_WMMA_F32_16X16X4_F32. CC is compile-time so column guards vanish when the
// grid tiles CC exactly. B fragments are reused across RT row tiles.
// ---------------------------------------------------------------------------
template <int RT, int CT, int CC, bool ACC>
__global__ __launch_bounds__(32) void gemm_wmma_f32(
    const float* __restrict__ A, const float* __restrict__ Bw,
    const int* __restrict__ gather, const float* __restrict__ bias,
    float* __restrict__ Out, int K) {
  constexpr bool GUARD = (CC % (16 * CT)) != 0;  // false for CC=256 tiles
  const int lane = threadIdx.x;      // 0..31
  const int l    = lane & 15;        // row (A) / col (B,C,D) within a tile
  const int half = lane >> 4;        // selects K pair (A/B) and M+8 (C/D)
  const int row0 = blockIdx.x * (16 * RT);
  const int n0   = blockIdx.y * (16 * CT);
  const int ko   = 2 * half;

  const float* __restrict__ Arow[RT];
#pragma unroll
  for (int rt = 0; rt < RT; ++rt) {
    const int r  = row0 + 16 * rt + l;
    const int ar = gather ? gather[r] : r;
    Arow[rt] = A + (size_t)ar * K;
  }

  const v8f zero = {0.f, 0.f, 0.f, 0.f, 0.f, 0.f, 0.f, 0.f};
  v8f acc[RT][CT];
#pragma unroll
  for (int rt = 0; rt < RT; ++rt)
#pragma unroll
    for (int t = 0; t < CT; ++t) acc[rt][t] = zero;

  const int K4 = K & ~3;
#pragma unroll 2
  for (int k = 0; k < K4; k += 4) {
    v2f a[RT];
#pragma unroll
    for (int rt = 0; rt < RT; ++rt) {
      a[rt].x = Arow[rt][k + ko];
      a[rt].y = Arow[rt][k + ko + 1];
    }
#pragma unroll
    for (int t = 0; t < CT; ++t) {
      const int n = n0 + 16 * t + l;
      v2f b = {0.f, 0.f};
      if (!GUARD || n < CC) {
        b.x = Bw[(size_t)(k + ko) * CC + n];
        b.y = Bw[(size_t)(k + ko + 1) * CC + n];
      }
#pragma unroll
      for (int rt = 0; rt < RT; ++rt)
        acc[rt][t] = __builtin_amdgcn_wmma_f32_16x16x4_f32(
            false, a[rt], false, b, (short)0, acc[rt][t], false, false);
    }
  }
  if (K4 < K) {  // zero-padded K tail (K=963 case)
    const int k = K4;
    v2f a[RT];
#pragma unroll
    for (int rt = 0; rt < RT; ++rt) {
      a[rt].x = (k + ko     < K) ? Arow[rt][k + ko]     : 0.f;
      a[rt].y = (k + ko + 1 < K) ? Arow[rt][k + ko + 1] : 0.f;
    }
#pragma unroll
    for (int t = 0; t < CT; ++t) {
      const int n = n0 + 16 * t + l;
      v2f b = {0.f, 0.f};
      if (!GUARD || n < CC) {
        if (k + ko     < K) b.x = Bw[(size_t)(k + ko) * CC + n];
        if (k + ko + 1 < K) b.y = Bw[(size_t)(k + ko + 1) * CC + n];
      }
#pragma unroll
      for (int rt = 0; rt < RT; ++rt)
        acc[rt][t] = __builtin_amdgcn_wmma_f32_16x16x4_f32(
            false, a[rt], false, b, (short)0, acc[rt][t], false, false);
    }
  }

#pragma unroll
  for (int t = 0; t < CT; ++t) {
    const int n = n0 + 16 * t + l;
    if (GUARD && n >= CC) continue;
    const float bv = bias ? bias[n] : 0.f;
#pragma unroll
    for (int rt = 0; rt < RT; ++rt) {
#pragma unroll
      for (int v = 0; v < 8; ++v) {
        const int row = row0 + 16 * rt + 8 * half + v;  // C/D: VGPR v -> M=v (+8)
        const size_t o = (size_t)row * CC + n;
        const float val = acc[rt][t][v] + bv;
        if (ACC) Out[o] += val; else Out[o] = val;
      }
    }
  }
}

// ---------------------------------------------------------------------------
// Graph helpers
// ---------------------------------------------------------------------------
__global__ void k_fill1(float* p, int n) {
  int i = blockIdx.x * blockDim.x + threadIdx.x;
  if (i < n) p[i] = 1.0f;  // self-loop contributes 1 to the degree
}

__global__ void k_degcount(const int* __restrict__ dst, float* deg, int E) {
  int e = blockIdx.x * blockDim.x + threadIdx.x;
  if (e < E) atomicAdd(&deg[dst[e]], 1.0f);
}

__global__ void k_rsqrt(float* p, int n) {
  int i = blockIdx.x * blockDim.x + threadIdx.x;
  if (i < n) p[i] = rsqrtf(fmaxf(p[i], 1.0f));
}

// agg[i,c] = dinv[i]^2 * t[i,c]   (self-loop edge, also initializes agg)
__global__ void k_aggself(const float* __restrict__ t, const float* __restrict__ dinv,
                          float* __restrict__ agg, int n, int C) {
  int idx = blockIdx.x * blockDim.x + threadIdx.x;
  if (idx >= n * C) return;
  int i = idx / C;
  float d = dinv[i];
  agg[idx] = d * d * t[idx];
}

// agg[dst,c] += dinv[src]*dinv[dst] * t[src,c]  (one thread per (edge, channel))
__global__ void k_aggedges(const int* __restrict__ src, const int* __restrict__ dst,
                           const float* __restrict__ t, const float* __restrict__ dinv,
                           float* agg, int E, int C) {
  int idx = blockIdx.x * blockDim.x + threadIdx.x;
  if (idx >= E * C) return;
  int e = idx / C;
  int c = idx - e * C;
  int s = src[e], d = dst[e];
  float nm = dinv[s] * dinv[d];
  atomicAdd(&agg[(size_t)d * C + c], nm * t[(size_t)s * C + c]);
}

__global__ void k_biasact(const float* __restrict__ agg, const float* __restrict__ bias,
                          float* __restrict__ out, int n, int C, int relu) {
  int idx = blockIdx.x * blockDim.x + threadIdx.x;
  if (idx >= n * C) return;
  int c = idx % C;
  float v = agg[idx] + bias[c];
  out[idx] = relu ? fmaxf(v, 0.0f) : v;
}

__global__ void k_residual(float* __restrict__ h, const float* __restrict__ y, int n) {
  int i = blockIdx.x * blockDim.x + threadIdx.x;
  if (i < n) h[i] = (h[i] + y[i]) * 0.5f;
}

// ---------------------------------------------------------------------------
// Orchestration
// ---------------------------------------------------------------------------
extern "C" void kernel_launch(void* const* d_in, const int* in_sizes, int n_in,
                              void* d_out, int out_size, void* d_ws, size_t ws_size,
                              hipStream_t stream) {
  (void)in_sizes; (void)n_in; (void)out_size; (void)ws_size;
  const float* x     = (const float*)d_in[0];
  const int*   edge  = (const int*)d_in[1];
  const int*   symm  = (const int*)d_in[2];
  const float* linW  = (const float*)d_in[3];
  const float* linb  = (const float*)d_in[4];
  const float* c1W   = (const float*)d_in[5];
  const float* c1b   = (const float*)d_in[6];
  const float* blkW  = (const float*)d_in[7];
  const float* blkWs = (const float*)d_in[8];
  const float* blkb  = (const float*)d_in[9];
  const float* c3W   = (const float*)d_in[10];
  const float* c3b   = (const float*)d_in[11];

  float* xs_out  = (float*)d_out;                                  // [B,N,CHID]
  float* out_out = xs_out + (size_t)NBATCH * NNODES * CHID;        // [B,N,COUT]

  const size_t SZ = (size_t)NNODES * CHID;
  float* t1   = (float*)d_ws;     // [N,CHID] scratch (GEMM result / y2)
  float* agg  = t1 + SZ;          // [N,CHID] aggregation accumulator
  float* ybuf = agg + SZ;         // [N,CHID] x0 / y1 buffer
  float* dinv = ybuf + SZ;        // [N] degree -> deg^-1/2

  const dim3 g4(NNODES / 32, CHID / 64);  // RT=2, CT=4 -> 32 x 64 per wave
  const dim3 g1(NNODES / 32, 1);          // RT=2, CT=1 for the 3-col head
  const int TPB = 256;
  const int gNC = (NNODES * CHID + TPB - 1) / TPB;
  const int gEC = (NEDGES * CHID + TPB - 1) / TPB;
  const int gN  = (NNODES + TPB - 1) / TPB;
  const int gE  = (NEDGES + TPB - 1) / TPB;
  const int gN3 = (NNODES * COUT + TPB - 1) / TPB;
  const int gE3 = (NEDGES * COUT + TPB - 1) / TPB;

  for (int b = 0; b < NBATCH; ++b) {
    const float* xb   = x + (size_t)b * NNODES * CIN;
    const int* srcb   = edge + (size_t)b * 2 * NEDGES;
    const int* dstb   = srcb + NEDGES;
    const int* symmb  = symm + (size_t)b * NNODES;
    float* hb = xs_out + (size_t)b * NNODES * CHID;
    float* ob = out_out + (size_t)b * NNODES * COUT;

    // x0 = x @ linW + linb  -> ybuf
    gemm_wmma_f32<2, 4, CHID, false><<<g4, 32, 0, stream>>>(
        xb, linW, nullptr, linb, ybuf, CIN);

    // degrees: deg = 1 + count(dst); dinv = rsqrt(max(deg,1))
    k_fill1<<<gN, TPB, 0, stream>>>(dinv, NNODES);
    k_degcount<<<gE, TPB, 0, stream>>>(dstb, dinv, NEDGES);
    k_rsqrt<<<gN, TPB, 0, stream>>>(dinv, NNODES);

    // conv1: h = relu(gcn(x0))
    gemm_wmma_f32<2, 4, CHID, false><<<g4, 32, 0, stream>>>(
        ybuf, c1W, nullptr, nullptr, t1, CHID);
    k_aggself<<<gNC, TPB, 0, stream>>>(t1, dinv, agg, NNODES, CHID);
    k_aggedges<<<gEC, TPB, 0, stream>>>(srcb, dstb, t1, dinv, agg, NEDGES, CHID);
    k_biasact<<<gNC, TPB, 0, stream>>>(agg, c1b, hb, NNODES, CHID, 1);

    for (int i = 0; i < NBLOCK; ++i) {
      const float* W0  = blkW  + (size_t)(i * 2 + 0) * CHID * CHID;
      const float* Ws0 = blkWs + (size_t)(i * 2 + 0) * CHID * CHID;
      const float* b0  = blkb  + (size_t)(i * 2 + 0) * CHID;
      const float* W1  = blkW  + (size_t)(i * 2 + 1) * CHID * CHID;
      const float* Ws1 = blkWs + (size_t)(i * 2 + 1) * CHID * CHID;
      const float* b1  = blkb  + (size_t)(i * 2 + 1) * CHID;

      // y1 = relu(gsn(h)): agg = norm-agg(h@W0) + h[symm]@Ws0; y1 = relu(agg+b0)
      gemm_wmma_f32<2, 4, CHID, false><<<g4, 32, 0, stream>>>(
          hb, W0, nullptr, nullptr, t1, CHID);
      k_aggself<<<gNC, TPB, 0, stream>>>(t1, dinv, agg, NNODES, CHID);
      k_aggedges<<<gEC, TPB, 0, stream>>>(srcb, dstb, t1, dinv, agg, NEDGES, CHID);
      gemm_wmma_f32<2, 4, CHID, true><<<g4, 32, 0, stream>>>(
          hb, Ws0, symmb, nullptr, agg, CHID);
      k_biasact<<<gNC, TPB, 0, stream>>>(agg, b0, ybuf, NNODES, CHID, 1);

      // y2 = relu(gsn(y1))  -> t1
      gemm_wmma_f32<2, 4, CHID, false><<<g4, 32, 0, stream>>>(
          ybuf, W1, nullptr, nullptr, t1, CHID);
      k_aggself<<<gNC, TPB, 0, stream>>>(t1, dinv, agg, NNODES, CHID);
      k_aggedges<<<gEC, TPB, 0, stream>>>(srcb, dstb, t1, dinv, agg, NEDGES, CHID);
      gemm_wmma_f32<2, 4, CHID, true><<<g4, 32, 0, stream>>>(
          ybuf, Ws1, symmb, nullptr, agg, CHID);
      k_biasact<<<gNC, TPB, 0, stream>>>(agg, b1, t1, NNODES, CHID, 1);

      // h = (h + y2) * 0.5
      k_residual<<<gNC, TPB, 0, stream>>>(hb, t1, NNODES * CHID);
    }

    // conv3 head: out = gcn(h, c3W, c3b)   (no relu)
    gemm_wmma_f32<2, 1, COUT, false><<<g1, 32, 0, stream>>>(
        hb, c3W, nullptr, nullptr, t1, CHID);
    k_aggself<<<gN3, TPB, 0, stream>>>(t1, dinv, agg, NNODES, COUT);
    k_aggedges<<<gE3, TPB, 0, stream>>>(srcb, dstb, t1, dinv, agg, NEDGES, COUT);
    k_biasact<<<gN3, TPB, 0, stream>>>(agg, c3b, ob, NNODES, COUT, 0);
  }
}